// BaseAttentionEntityPooler_17557826306583
// MI455X (gfx1250) — compile-verified
//
#include <hip/hip_runtime.h>
#include <hip/hip_bf16.h>
#include <math.h>

// Problem dimensions (from reference): F=1, B=16, L=4096, D=2048, T=4
#define BATCH 16
#define SEQL  4096
#define DIM   2048
#define NSPAN 4
#define MAXACT 64   // T * MAX_SPAN upper bound on active positions per batch

typedef __attribute__((ext_vector_type(2))) float v2f;
typedef __attribute__((ext_vector_type(8))) float v8f;

__device__ inline float wave_reduce_sum(float v) {
    // wave32 butterfly reduction
    v += __shfl_xor(v, 16, 32);
    v += __shfl_xor(v,  8, 32);
    v += __shfl_xor(v,  4, 32);
    v += __shfl_xor(v,  2, 32);
    v += __shfl_xor(v,  1, 32);
    return v;
}

// ---------------------------------------------------------------------------
// Kernel 1: one block per batch element.
//   - builds span mask bitmap (dedups overlapping spans)
//   - zeroes the attn output row
//   - computes scores only for active positions (sparse!), softmax, probs
//   - accumulates pooled[b, :] into workspace
// ---------------------------------------------------------------------------
__global__ __launch_bounds__(256) void span_pool_kernel(
    const float* __restrict__ hidden,          // [B, L, D]
    const int*   __restrict__ token_idxs,      // [1, B, T, 2]
    const float* __restrict__ pooled_entities, // [B, D]
    const float* __restrict__ W_align,         // [1, 2D]
    const float* __restrict__ b_align,         // [1]
    float*       __restrict__ attn_out,        // [B, L]
    float*       __restrict__ pooled_ws)       // [B, D]
{
    const int b    = blockIdx.x;
    const int tid  = threadIdx.x;
    const int lane = tid & 31;
    const int wid  = tid >> 5;        // 8 waves

    __shared__ unsigned mask_bits[SEQL / 32];  // 128 words
    __shared__ int   npos;
    __shared__ int   pos[MAXACT];
    __shared__ float sc[MAXACT];
    __shared__ float wsum[8];
    __shared__ float cb_sh, max_sh, sum_sh;

    // ---- init bitmap + counter ----
    if (tid < SEQL / 32) mask_bits[tid] = 0u;
    if (tid == 0) npos = 0;
    __syncthreads();

    // ---- mark span positions (thread = (span, offset) pair) ----
    if (tid < NSPAN * 16) {
        const int t   = tid >> 4;
        const int off = tid & 15;
        const int base  = (b * NSPAN + t) * 2;
        const int start = token_idxs[base + 0];
        const int end   = token_idxs[base + 1];   // exclusive
        const int l = start + off;
        if (l < end && l >= 0 && l < SEQL)
            atomicOr(&mask_bits[l >> 5], 1u << (l & 31));
    }

    // ---- zero this batch's attention row ----
    for (int l = tid; l < SEQL; l += 256)
        attn_out[b * SEQL + l] = 0.0f;
    __syncthreads();

    // ---- enumerate active positions ----
    if (tid < SEQL / 32) {
        unsigned w = mask_bits[tid];
        while (w) {
            const int bit = __ffs(w) - 1;
            w &= w - 1;
            const int p = atomicAdd(&npos, 1);
            if (p < MAXACT) pos[p] = (tid << 5) | bit;
        }
    }

    // ---- constant entity-alignment term (same for every active l) ----
    {
        float part = 0.0f;
        const float* pe = pooled_entities + b * DIM;
        for (int d = tid; d < DIM; d += 256)
            part += pe[d] * W_align[d];
        part = wave_reduce_sum(part);
        if (lane == 0) wsum[wid] = part;
    }
    __syncthreads();
    if (tid == 0) {
        float c = b_align[0];
        #pragma unroll
        for (int i = 0; i < 8; ++i) c += wsum[i];
        cb_sh = c;
        if (npos > MAXACT) npos = MAXACT;
    }
    __syncthreads();

    const int n = npos;

    // ---- scores: one wave per active position, lanes stride over D ----
    for (int i = wid; i < n; i += 8) {
        const float* hrow = hidden + ((size_t)b * SEQL + pos[i]) * DIM;
        const float* wa   = W_align + DIM;
        float p = 0.0f;
        for (int d = lane; d < DIM; d += 32)
            p += hrow[d] * wa[d];
        p = wave_reduce_sum(p);
        if (lane == 0) sc[i] = p + cb_sh;
    }
    __syncthreads();

    // ---- softmax over the (tiny) active set ----
    if (tid == 0) {
        float m = -INFINITY;
        for (int i = 0; i < n; ++i) m = fmaxf(m, sc[i]);
        float s = 0.0f;
        for (int i = 0; i < n; ++i) s += expf(sc[i] - m);
        max_sh = m;
        sum_sh = s;
    }
    __syncthreads();
    for (int i = tid; i < n; i += 256) {
        const float p = expf(sc[i] - max_sh) / sum_sh;
        sc[i] = p;
        attn_out[b * SEQL + pos[i]] = p;
    }
    __syncthreads();

    // ---- pooled[b, d] = sum over active l of prob * hidden[b, l, d] ----
    for (int d = tid; d < DIM; d += 256) {
        float acc = 0.0f;
        for (int i = 0; i < n; ++i)
            acc += sc[i] * hidden[((size_t)b * SEQL + pos[i]) * DIM + d];
        pooled_ws[b * DIM + d] = acc;
    }
}

// ---------------------------------------------------------------------------
// Kernel 2: projected = tanh(pooled @ W_out^T + b_out) via V_WMMA_F32_16X16X4_F32.
// One wave per 16x16 output tile; M=16 = batch, N tiles over 2048, K step 4.
//
// A (16x4 f32) per-lane layout (ISA 7.12.2): lanes 0-15 hold {K=0,K=1} for
// M=lane; lanes 16-31 hold {K=2,K=3} for M=lane-16.  B (4x16) mirrors this
// with N=lane%16.  C/D: VGPR v holds M=v (lanes 0-15) / M=v+8 (lanes 16-31),
// N = lane%16.
// ---------------------------------------------------------------------------
__global__ __launch_bounds__(32) void out_proj_wmma_kernel(
    const float* __restrict__ pooled,  // [16, 2048]  (A, row-major)
    const float* __restrict__ W_out,   // [2048, 2048] (B[k][n] = W_out[n][k])
    const float* __restrict__ b_out,   // [2048]
    float*       __restrict__ proj)    // [16, 2048]
{
    const int lane = threadIdx.x;        // 0..31
    const int tile = blockIdx.x;         // 0..127
    const int half = lane >> 4;          // K sub-block select
    const int l16  = lane & 15;

    const float* Arow = pooled + l16 * DIM;            // A row M=l16
    const float* Brow = W_out + (size_t)(tile * 16 + l16) * DIM; // B col N

    v8f c = {0.f, 0.f, 0.f, 0.f, 0.f, 0.f, 0.f, 0.f};

    #pragma unroll 4
    for (int k = 0; k < DIM; k += 4) {
        const int ka = k + half * 2;                   // even -> 8B aligned
        const v2f a  = *(const v2f*)(Arow + ka);
        const v2f bb = *(const v2f*)(Brow + ka);
        c = __builtin_amdgcn_wmma_f32_16x16x4_f32(
                /*neg_a=*/false, a, /*neg_b=*/false, bb,
                /*c_mod=*/(short)0, c, /*reuse_a=*/false, /*reuse_b=*/false);
    }

    const int ncol = tile * 16 + l16;
    const float bias = b_out[ncol];
    #pragma unroll
    for (int v = 0; v < 8; ++v) {
        const int m = v + half * 8;
        proj[m * DIM + ncol] = tanhf(c[v] + bias);
    }
}

// ---------------------------------------------------------------------------
extern "C" void kernel_launch(void* const* d_in, const int* in_sizes, int n_in,
                              void* d_out, int out_size, void* d_ws, size_t ws_size,
                              hipStream_t stream) {
    const float* hidden          = (const float*)d_in[0]; // [B,L,D]
    const int*   token_idxs      = (const int*)  d_in[1]; // [1,B,T,2]
    const float* pooled_entities = (const float*)d_in[2]; // [B,D]
    const float* W_align         = (const float*)d_in[3]; // [1,2D]
    const float* b_align         = (const float*)d_in[4]; // [1]
    const float* W_out           = (const float*)d_in[5]; // [D,D]
    const float* b_out           = (const float*)d_in[6]; // [D]

    float* out  = (float*)d_out;
    float* proj = out;                  // [B, D]   first output
    float* attn = out + BATCH * DIM;    // [1, B, L, 1] second output
    float* pooled_ws = (float*)d_ws;    // [B, D] scratch (128 KB)

    span_pool_kernel<<<BATCH, 256, 0, stream>>>(
        hidden, token_idxs, pooled_entities, W_align, b_align, attn, pooled_ws);

    out_proj_wmma_kernel<<<DIM / 16, 32, 0, stream>>>(
        pooled_ws, W_out, b_out, proj);
}